// CTCLoss_89094801588412
// MI455X (gfx1250) — compile-verified
//
#include <hip/hip_runtime.h>
#include <math.h>

// Problem dims (fixed by the reference's setup_inputs)
#define BB 64
#define TT 256
#define VV 6625
#define SS 25
#define SX 51
#define NEGV (-1e30f)

__device__ __forceinline__ float lse3(float a, float b, float c) {
  float m = fmaxf(a, fmaxf(b, c));
  return m + __logf(__expf(a - m) + __expf(b - m) + __expf(c - m));
}

// ---------------------------------------------------------------------------
// Phase 1: per (b,t) row. Stage the 6625-float row into LDS with fully
// unrolled async b32 issues (single HBM pass), pull 26 values/thread into
// registers, compute lse = max + log(sum exp(x-max)), then gather the 26
// needed emission log-probs from LDS: emit26[row][j] = pred[row][cls] - lse.
// ---------------------------------------------------------------------------
__global__ __launch_bounds__(256) void lse_gather_kernel(
    const float* __restrict__ pred, const int* __restrict__ target,
    float* __restrict__ emit26) {
  __shared__ float sbuf[VV];
  __shared__ float sred[8];

  const int row = blockIdx.x;          // row = b*TT + t
  const int tid = threadIdx.x;
  const int lane = tid & 31;
  const int wid = tid >> 5;
  const float* rowp = pred + (size_t)row * VV;

  // --- async stage: global -> LDS; IOFFSET applies to both addresses, so
  //     one address pair + 25 unrolled issues with immediate offsets. ---
  unsigned lds_base = (unsigned)(unsigned long long)(&sbuf[0]);
  unsigned laddr = lds_base + (unsigned)tid * 4u;
  const float* g = rowp + tid;
#pragma unroll
  for (int k = 0; k < 25; ++k) {
    asm volatile("global_load_async_to_lds_b32 %0, %1, off offset:%2"
                 :: "v"(laddr), "v"(g), "i"(k * 1024) : "memory");
  }
  if (tid < (VV - 25 * 256)) {  // tail: 225 threads
    asm volatile("global_load_async_to_lds_b32 %0, %1, off offset:%2"
                 :: "v"(laddr), "v"(g), "i"(25 * 1024) : "memory");
  }
  asm volatile("s_wait_asynccnt 0" ::: "memory");
  __syncthreads();

  // --- one LDS read pass into registers; both reductions run from regs ---
  float x[26];
#pragma unroll
  for (int k = 0; k < 25; ++k) x[k] = sbuf[tid + (k << 8)];
  x[25] = (tid < (VV - 25 * 256)) ? sbuf[tid + 6400] : -3.0e38f;

  float m = x[0];
#pragma unroll
  for (int k = 1; k < 26; ++k) m = fmaxf(m, x[k]);
#pragma unroll
  for (int off = 16; off; off >>= 1) m = fmaxf(m, __shfl_xor(m, off, 32));
  if (lane == 0) sred[wid] = m;
  __syncthreads();
  float bm = sred[0];
#pragma unroll
  for (int w = 1; w < 8; ++w) bm = fmaxf(bm, sred[w]);
  __syncthreads();  // protect sred before reuse

  float s = 0.0f;
#pragma unroll
  for (int k = 0; k < 26; ++k) s += __expf(x[k] - bm);  // pad exp -> 0
#pragma unroll
  for (int off = 16; off; off >>= 1) s += __shfl_xor(s, off, 32);
  if (lane == 0) sred[wid] = s;
  __syncthreads();
  float bs = 0.0f;
#pragma unroll
  for (int w = 0; w < 8; ++w) bs += sred[w];
  float lse = bm + logf(bs);

  // --- gather the 26 classes that matter (blank + 25 labels), from LDS ---
  if (tid < 26) {
    const int b = row >> 8;  // TT == 256
    int cls = (tid == 0) ? 0 : target[b * SS + (tid - 1)];
    emit26[(size_t)row * 26 + tid] = sbuf[cls] - lse;
  }
}

// ---------------------------------------------------------------------------
// Phase 2: CTC forward DP. One wave32 per batch sample; lane l holds states
// l and 32+l (Sx=51). The sample's whole 26624-B emission table is bulk-
// staged to LDS with async b128 (52 x 512B issues), so the 255-step serial
// chain only touches LDS (+ one-step register prefetch). No barriers.
// ---------------------------------------------------------------------------
__global__ __launch_bounds__(32) void ctc_dp_kernel(
    const float* __restrict__ emit26, const int* __restrict__ target,
    const int* __restrict__ length, float* __restrict__ per_sample) {
  __shared__ float se[TT * 26 + 32];   // +1 padded row for unconditional prefetch
  const int b = blockIdx.x;
  const int l = threadIdx.x;  // lane 0..31
  const float* e = emit26 + (size_t)b * TT * 26;
  const int* tg = target + b * SS;

  // --- async bulk stage: 26624 B = 52 issues x (32 lanes x 16 B) ---
  unsigned lds_base = (unsigned)(unsigned long long)(&se[0]);
  unsigned laddr = lds_base + (unsigned)l * 16u;
  const float* g = e + l * 4;
#pragma unroll
  for (int k = 0; k < 52; ++k) {
    asm volatile("global_load_async_to_lds_b128 %0, %1, off offset:%2"
                 :: "v"(laddr), "v"(g), "i"(k * 512) : "memory");
  }
  asm volatile("s_wait_asynccnt 0" ::: "memory");
  __syncthreads();  // single wave; cheap

  const int sL = l;        // low state
  const int sH = 32 + l;   // high state, valid if <= 50
  const bool vH = (sH < SX);

  // emission column: even state -> blank(0), odd -> 1+j
  const int eL = (sL & 1) ? (1 + ((sL - 1) >> 1)) : 0;
  const int eH = (vH && (sH & 1)) ? (1 + ((sH - 1) >> 1)) : 0;

  // skip (s-2 transition) flags
  bool skL = false, skH = false;
  if (sL & 1) { int j = (sL - 1) >> 1; skL = (j == 0) || (tg[j] != tg[j - 1]); }
  if (vH && (sH & 1)) { int j = (sH - 1) >> 1; skH = (j == 0) || (tg[j] != tg[j - 1]); }

  // t = 0 init + prefetch t = 1 emissions
  float aL = (sL < 2) ? se[eL] : NEGV;
  float aH = NEGV;
  float emL = se[26 + eL];
  float emH = se[26 + eH];

  for (int t = 1; t < TT; ++t) {
    // prefetch next step's emissions (padded row makes t=255 read safe)
    float nemL = se[(t + 1) * 26 + eL];
    float nemH = se[(t + 1) * 26 + eH];

    // predecessors via wave shuffles (read old alpha before updating)
    float p1L = __shfl_up(aL, 1, 32); if (l == 0) p1L = NEGV;
    float p2L = __shfl_up(aL, 2, 32); if (l < 2) p2L = NEGV;
    float a31 = __shfl(aL, 31, 32);
    float a30 = __shfl(aL, 30, 32);
    float p1H = __shfl_up(aH, 1, 32); if (l == 0) p1H = a31;
    float p2H = __shfl_up(aH, 2, 32); if (l == 0) p2H = a30; else if (l == 1) p2H = a31;

    float nL = lse3(aL, p1L, skL ? p2L : NEGV) + emL;
    float nH = lse3(aH, p1H, skH ? p2H : NEGV) + emH;
    aL = nL;
    aH = vH ? nH : NEGV;
    emL = nemL;
    emH = nemH;
  }

  // final: ll = logsumexp(alpha[2L-1], alpha[2L]); loss = -ll (zero_infinity)
  const int len = length[b];
  const int i2 = 2 * len;           // 2..50 (uniform across the wave)
  const int s1 = i2 - 1, s2 = i2;
  float v1 = (s1 < 32) ? __shfl(aL, s1, 32) : __shfl(aH, s1 - 32, 32);
  float v2 = (s2 < 32) ? __shfl(aL, s2, 32) : __shfl(aH, s2 - 32, 32);
  float mm = fmaxf(v1, v2);
  float ll = mm + logf(__expf(v1 - mm) + __expf(v2 - mm));
  float loss = -ll;
  if (!isfinite(loss)) loss = 0.0f;
  if (l == 0) per_sample[b] = loss / (float)len;
}

// ---------------------------------------------------------------------------
// Phase 3: deterministic mean over 64 per-sample losses (no float atomics).
// ---------------------------------------------------------------------------
__global__ __launch_bounds__(32) void finalize_kernel(
    const float* __restrict__ per_sample, float* __restrict__ out) {
  const int l = threadIdx.x;
  float v = per_sample[l] + per_sample[l + 32];
#pragma unroll
  for (int off = 16; off; off >>= 1) v += __shfl_xor(v, off, 32);
  if (l == 0) out[0] = v / (float)BB;
}

extern "C" void kernel_launch(void* const* d_in, const int* in_sizes, int n_in,
                              void* d_out, int out_size, void* d_ws, size_t ws_size,
                              hipStream_t stream) {
  (void)in_sizes; (void)n_in; (void)out_size; (void)ws_size;
  const float* pred   = (const float*)d_in[0];  // [B,T,V] f32
  const int*   target = (const int*)d_in[1];    // [B,S] i32
  const int*   length = (const int*)d_in[2];    // [B] i32
  float* out = (float*)d_out;                   // scalar f32

  float* emit26     = (float*)d_ws;                       // [B*T, 26]
  float* per_sample = emit26 + (size_t)BB * TT * 26;      // [B]

  lse_gather_kernel<<<BB * TT, 256, 0, stream>>>(pred, target, emit26);
  ctc_dp_kernel<<<BB, 32, 0, stream>>>(emit26, target, length, per_sample);
  finalize_kernel<<<1, 32, 0, stream>>>(per_sample, out);
}